// GCLayer_83373905150287
// MI455X (gfx1250) — compile-verified
//
#include <hip/hip_runtime.h>
#include <hip/hip_bf16.h>

typedef __attribute__((ext_vector_type(2))) float v2f;
typedef __attribute__((ext_vector_type(8))) float v8f;

__device__ __forceinline__ v8f wmma4(v2f a, v2f b, v8f c) {
    // D = A(16x4 f32) * B(4x16 f32) + C(16x16 f32)
    return __builtin_amdgcn_wmma_f32_16x16x4_f32(false, a, false, b, (short)0, c,
                                                 false, false);
}

__device__ __forceinline__ float sigmoidf_(float x) { return 1.0f / (1.0f + expf(-x)); }
__device__ __forceinline__ float siluf_(float x)    { return x * sigmoidf_(x); }

template <bool ALIGNED>
__device__ __forceinline__ v2f ldpair(const float* __restrict__ p) {
    if constexpr (ALIGNED) {
        return *(const v2f*)p;          // global_load_b64
    } else {
        v2f b; b.x = p[0]; b.y = p[1];  // odd stride (257): scalar pair
        return b;
    }
}

// ---------------------------------------------------------------------------
// fp32-WMMA GEMM: out[i][n] = alpha * sum_k in[i][k] * W[n*W_LD + w_off + k]
//                             + bias[n] + resid[i][n]
// One wave / 16-row tile. Register-tiled: 16x128 of C lives in 8 v8f
// accumulators; one pass over K; per k-step 1 A-frag + 8 B-frags feed 8
// independent WMMA chains, with depth-1 rotation for latency hiding.
// ---------------------------------------------------------------------------
template <int KSTEPS, int W_LD>
__launch_bounds__(32, 1)
__global__ void gemm16(const float* __restrict__ in, int in_ld,
                       const float* __restrict__ W, int w_off,
                       const float* __restrict__ bias,
                       const float* __restrict__ resid, int resid_ld,
                       float* __restrict__ out, int out_ld,
                       int outc, int nrows, float alpha) {
    constexpr bool WAL = (W_LD % 2 == 0);
    const int lane = threadIdx.x & 31;
    const int m    = lane & 15;      // A-row (M) / B,C col (N)
    const int kh   = lane >> 4;      // K-half selector
    const int row0 = blockIdx.x * 16;

    int ar = row0 + m;
    if (ar >= nrows) ar = nrows - 1;
    const float* __restrict__ inrow = in + (size_t)ar * in_ld + 2 * kh;

    const int nChunks = outc >> 7;                   // 128 output cols per chunk
    for (int ch = 0; ch < nChunks; ++ch) {
        asm volatile("" ::: "memory");               // block cross-chunk hoisting
        const int nbase = ch * 128 + m;
        const float* __restrict__ wbase = W + (size_t)nbase * W_LD + w_off + 2 * kh;

        v8f acc[8];
#pragma unroll
        for (int j = 0; j < 8; ++j) acc[j] = (v8f){0.f,0.f,0.f,0.f,0.f,0.f,0.f,0.f};

        v2f acur = *(const v2f*)(inrow);
        v2f bcur[8];
#pragma unroll
        for (int j = 0; j < 8; ++j) bcur[j] = ldpair<WAL>(wbase + j * 16 * W_LD);

        for (int kk = 0; kk < KSTEPS - 1; ++kk) {
            v2f anxt = *(const v2f*)(inrow + 4 * (kk + 1));
            v2f bnxt[8];
#pragma unroll
            for (int j = 0; j < 8; ++j)
                bnxt[j] = ldpair<WAL>(wbase + j * 16 * W_LD + 4 * (kk + 1));
#pragma unroll
            for (int j = 0; j < 8; ++j) acc[j] = wmma4(acur, bcur[j], acc[j]);
            acur = anxt;
#pragma unroll
            for (int j = 0; j < 8; ++j) bcur[j] = bnxt[j];
        }
#pragma unroll
        for (int j = 0; j < 8; ++j) acc[j] = wmma4(acur, bcur[j], acc[j]);

        // epilogue
#pragma unroll
        for (int j = 0; j < 8; ++j) {
            const int n  = ch * 128 + 16 * j + m;
            const float bv = bias ? bias[n] : 0.f;
#pragma unroll
            for (int v = 0; v < 8; ++v) {
                const int r = row0 + v + 8 * kh;     // C/D row: M = v + 8*kh
                if (r < nrows) {
                    float o = alpha * acc[j][v] + bv;
                    if (resid) o += resid[(size_t)r * resid_ld + n];
                    out[(size_t)r * out_ld + n] = o;
                }
            }
        }
    }
}

// ---------------------------------------------------------------------------
// Edge kernel: 2 waves / block, each wave owns one 16-edge tile.
//   a1 = silu(P[row] + Q[col] + (dist*emask)*w1c + b1)        -> LDS [16x256]
//   a2 = silu(a1 @ att_w2^T + b2)  (fp32 WMMA, K=256, 8 acc chains, 1 K pass)
//   att = sigmoid(a2 . att_w3 + b3) * emask
//   agg[row] += (att/NORM) * x[col]        (global_atomic_add_f32)
// ---------------------------------------------------------------------------
__launch_bounds__(64, 1)
__global__ void edge_kernel(const float* __restrict__ P, const float* __restrict__ Q,
                            const float* __restrict__ x,
                            const float* __restrict__ distances,
                            const float* __restrict__ edge_mask,
                            const float* __restrict__ att_w1,  // [256 x 257]
                            const float* __restrict__ att_b1,  // [256]
                            const float* __restrict__ att_w2,  // [128 x 256]
                            const float* __restrict__ att_b2,  // [128]
                            const float* __restrict__ att_w3,  // [128]
                            const float* __restrict__ att_b3,  // [1]
                            const int* __restrict__ edges,     // [2 x E]
                            float* __restrict__ agg,
                            int E, float inv_norm) {
    __shared__ float a1s[2][16][260];   // 260: rows 16B-aligned, banks 4m+c
    __shared__ float a2s[2][16][130];
    __shared__ float w1cs[256], b1s[256];
    __shared__ float attv[2][16];

    const int lane = threadIdx.x & 31;
    const int w    = threadIdx.x >> 5;
    const int e0   = (blockIdx.x * 2 + w) * 16;
    const int* __restrict__ rowIdx = edges;
    const int* __restrict__ colIdx = edges + E;

    // stage last column of att_w1 (stride-257 gather, once per block) + b1
    for (int t = threadIdx.x; t < 256; t += blockDim.x) {
        w1cs[t] = att_w1[t * 257 + 256];
        b1s[t]  = att_b1[t];
    }
    __syncthreads();

    // ---- layer 1 (factored): build a1 tile in LDS, float4 traffic ----
    for (int mm = 0; mm < 16; ++mm) {
        int e = e0 + mm;
        if (e >= E) e = E - 1;                       // clamped read; store guarded later
        const int r = rowIdx[e], c = colIdx[e];
        const float dterm = distances[e] * edge_mask[e];
        const float* __restrict__ Pr = P + (size_t)r * 256;
        const float* __restrict__ Qc = Q + (size_t)c * 256;
#pragma unroll
        for (int j = 0; j < 2; ++j) {
            const int f0 = lane * 4 + 128 * j;
            const float4 p = *(const float4*)(Pr + f0);
            const float4 q = *(const float4*)(Qc + f0);
            float4 o;
            o.x = siluf_(p.x + q.x + dterm * w1cs[f0 + 0] + b1s[f0 + 0]);
            o.y = siluf_(p.y + q.y + dterm * w1cs[f0 + 1] + b1s[f0 + 1]);
            o.z = siluf_(p.z + q.z + dterm * w1cs[f0 + 2] + b1s[f0 + 2]);
            o.w = siluf_(p.w + q.w + dterm * w1cs[f0 + 3] + b1s[f0 + 3]);
            *(float4*)&a1s[w][mm][f0] = o;
        }
    }
    __syncthreads();

    const int m  = lane & 15;
    const int kh = lane >> 4;

    // ---- layer 2: [16x256] @ att_w2^T -> [16x128], register-tiled WMMA ----
    {
        const float* __restrict__ wbase = att_w2 + (size_t)m * 256 + 2 * kh;
        const float* __restrict__ arow  = &a1s[w][m][2 * kh];

        v8f acc[8];
#pragma unroll
        for (int j = 0; j < 8; ++j) acc[j] = (v8f){0.f,0.f,0.f,0.f,0.f,0.f,0.f,0.f};

        v2f acur = *(const v2f*)(arow);              // ds_load_b64
        v2f bcur[8];
#pragma unroll
        for (int j = 0; j < 8; ++j) bcur[j] = *(const v2f*)(wbase + j * 16 * 256);

        for (int kk = 0; kk < 63; ++kk) {
            v2f anxt = *(const v2f*)(arow + 4 * (kk + 1));
            v2f bnxt[8];
#pragma unroll
            for (int j = 0; j < 8; ++j)
                bnxt[j] = *(const v2f*)(wbase + j * 16 * 256 + 4 * (kk + 1));
#pragma unroll
            for (int j = 0; j < 8; ++j) acc[j] = wmma4(acur, bcur[j], acc[j]);
            acur = anxt;
#pragma unroll
            for (int j = 0; j < 8; ++j) bcur[j] = bnxt[j];
        }
#pragma unroll
        for (int j = 0; j < 8; ++j) acc[j] = wmma4(acur, bcur[j], acc[j]);

#pragma unroll
        for (int j = 0; j < 8; ++j) {
            const int n  = 16 * j + m;
            const float bv = att_b2[n];
#pragma unroll
            for (int v = 0; v < 8; ++v) {
                const int M = v + 8 * kh;            // edge-row inside the tile
                a2s[w][M][n] = siluf_(acc[j][v] + bv);
            }
        }
    }
    __syncthreads();

    // ---- layer 3: per-edge dot + sigmoid ----
    float part = 0.f;
    const int fbase = 64 * kh;
#pragma unroll
    for (int f = 0; f < 64; ++f) part += a2s[w][m][fbase + f] * att_w3[fbase + f];
    part += __shfl_xor(part, 16, 32);
    if (kh == 0) {
        int e = e0 + m;
        if (e >= E) e = E - 1;
        attv[w][m] = sigmoidf_(part + att_b3[0]) * edge_mask[e];
    }
    __syncthreads();

    // ---- aggregate: agg[row] += (att * 1/NORM) * x[col] ----
    for (int mm = 0; mm < 16; ++mm) {
        const int e = e0 + mm;
        if (e >= E) break;
        const int r = rowIdx[e], c = colIdx[e];
        const float am = attv[w][mm] * inv_norm;
        const float4 xv = *(const float4*)(x + (size_t)c * 128 + lane * 4);
        float* __restrict__ ar = agg + (size_t)r * 128 + lane * 4;
        atomicAdd(&ar[0], am * xv.x);
        atomicAdd(&ar[1], am * xv.y);
        atomicAdd(&ar[2], am * xv.z);
        atomicAdd(&ar[3], am * xv.w);
    }
}

// ---------------------------------------------------------------------------
// Row-wise LayerNorm (+SiLU), one wave per 128-wide row.
// ---------------------------------------------------------------------------
__global__ void ln_silu(const float* __restrict__ in, const float* __restrict__ g,
                        const float* __restrict__ b, float* __restrict__ out,
                        int nrows) {
    const int lane = threadIdx.x & 31;
    const int w    = threadIdx.x >> 5;
    const int row  = blockIdx.x * (blockDim.x >> 5) + w;
    if (row >= nrows) return;
    const float* __restrict__ xr = in + (size_t)row * 128;
    float v[4], s = 0.f, sq = 0.f;
#pragma unroll
    for (int i = 0; i < 4; ++i) {
        v[i] = xr[lane + 32 * i];
        s += v[i];
        sq += v[i] * v[i];
    }
#pragma unroll
    for (int off = 16; off >= 1; off >>= 1) {
        s  += __shfl_xor(s,  off, 32);
        sq += __shfl_xor(sq, off, 32);
    }
    const float mean = s * (1.0f / 128.0f);
    const float var  = sq * (1.0f / 128.0f) - mean * mean;
    const float rstd = rsqrtf(var + 1e-5f);
    float* __restrict__ orow = out + (size_t)row * 128;
#pragma unroll
    for (int i = 0; i < 4; ++i) {
        const int f = lane + 32 * i;
        const float y = (v[i] - mean) * rstd * g[f] + b[f];
        orow[f] = siluf_(y);
    }
}

// ---------------------------------------------------------------------------
extern "C" void kernel_launch(void* const* d_in, const int* in_sizes, int n_in,
                              void* d_out, int out_size, void* d_ws, size_t ws_size,
                              hipStream_t stream) {
    const float* h         = (const float*)d_in[0];
    const float* distances = (const float*)d_in[1];
    /* node_mask d_in[2] unused by the reference */
    const float* edge_mask = (const float*)d_in[3];
    const float* lin_w     = (const float*)d_in[4];
    const float* lin_b     = (const float*)d_in[5];
    const float* att_w1    = (const float*)d_in[6];
    const float* att_b1    = (const float*)d_in[7];
    const float* att_w2    = (const float*)d_in[8];
    const float* att_b2    = (const float*)d_in[9];
    const float* att_w3    = (const float*)d_in[10];
    const float* att_b3    = (const float*)d_in[11];
    const float* nm_w1     = (const float*)d_in[12];
    const float* nm_b1     = (const float*)d_in[13];
    const float* nm_ln_g   = (const float*)d_in[14];
    const float* nm_ln_b   = (const float*)d_in[15];
    const float* nm_w2     = (const float*)d_in[16];
    const float* nm_b2     = (const float*)d_in[17];
    const float* ln_g      = (const float*)d_in[18];
    const float* ln_b      = (const float*)d_in[19];
    const int*   edges     = (const int*)d_in[20];

    const int N = in_sizes[0] / 128;
    const int E = in_sizes[1];

    float* ws   = (float*)d_ws;
    float* xbuf = ws;                                // N*128
    float* Pbuf = xbuf + (size_t)N * 128;            // N*256
    float* Qbuf = Pbuf + (size_t)N * 256;            // N*256
    float* abuf = Qbuf + (size_t)N * 256;            // N*128 (agg, later reused as hh)
    float* tbuf = abuf + (size_t)N * 128;            // N*128

    const int rowTiles = (N + 15) / 16;

    // x = h @ lin_w^T + lin_b
    gemm16<32, 128><<<rowTiles, 32, 0, stream>>>(h, 128, lin_w, 0, lin_b,
                                                 nullptr, 0, xbuf, 128, 128, N, 1.0f);
    // P = x @ att_w1[:, :128]^T ; Q = x @ att_w1[:, 128:256]^T   (odd stride 257)
    gemm16<32, 257><<<rowTiles, 32, 0, stream>>>(xbuf, 128, att_w1, 0, nullptr,
                                                 nullptr, 0, Pbuf, 256, 256, N, 1.0f);
    gemm16<32, 257><<<rowTiles, 32, 0, stream>>>(xbuf, 128, att_w1, 128, nullptr,
                                                 nullptr, 0, Qbuf, 256, 256, N, 1.0f);

    hipMemsetAsync(abuf, 0, (size_t)N * 128 * sizeof(float), stream);

    // edge phase: attention MLP + scaled segment-sum via fp32 atomics
    const int edgeBlocks = (E + 31) / 32;            // 2 tiles of 16 edges per block
    edge_kernel<<<edgeBlocks, 64, 0, stream>>>(Pbuf, Qbuf, xbuf, distances, edge_mask,
                                               att_w1, att_b1, att_w2, att_b2,
                                               att_w3, att_b3, edges, abuf,
                                               E, 1.0f / 100.0f);

    // node MLP: t = agg @ nm_w1^T + nm_b1 ; t = silu(LN(t))
    gemm16<32, 128><<<rowTiles, 32, 0, stream>>>(abuf, 128, nm_w1, 0, nm_b1,
                                                 nullptr, 0, tbuf, 128, 128, N, 1.0f);
    ln_silu<<<(N + 7) / 8, 256, 0, stream>>>(tbuf, nm_ln_g, nm_ln_b, tbuf, N);

    // hh = t @ nm_w2^T + nm_b2 + x   (write into abuf, agg no longer needed)
    gemm16<32, 128><<<rowTiles, 32, 0, stream>>>(tbuf, 128, nm_w2, 0, nm_b2,
                                                 xbuf, 128, abuf, 128, 128, N, 1.0f);

    // out = silu(LN(hh))
    ln_silu<<<(N + 7) / 8, 256, 0, stream>>>(abuf, ln_g, ln_b, (float*)d_out, N);
}